// LSTMAttentionDecoder_91250875171347
// MI455X (gfx1250) — compile-verified
//
#include <hip/hip_runtime.h>
#include <hip/hip_bf16.h>

typedef __attribute__((ext_vector_type(2))) float v2f;
typedef __attribute__((ext_vector_type(4))) float v4f;
typedef __attribute__((ext_vector_type(8))) float v8f;

#define V_SIZE 50257
#define E_SIZE 1024
#define H_SIZE 1024
#define S_SIZE 4096

// Workspace layout (float offsets). Total ~130.5K floats (~522 KB).
#define WS_X        0        // 1024   relu(emb[idx])
#define WS_GATES    1024     // 4096
#define WS_HNEW     5120     // 1024
#define WS_CNEW     6144     // 1024
#define WS_ASCORE   7168     // 4096   attention scores a[s]
#define WS_CTXPART  11264    // 64*1024 partial ctx sums
#define WS_CAT      76800    // 2048   [ctx ; h_new]
#define WS_FINAL    78848    // 1024   final_hidden
#define WS_MID      79872    // 50257  logits before log_softmax
#define WS_STATS    130304   // 2*64   per-block (max, sumexp)
#define WS_LOGZ     130432   // 1

__device__ __forceinline__ float sigmoidf_(float x) { return 1.0f / (1.0f + expf(-x)); }

// ---------------------------------------------------------------------------
// Wave-level fp32 matvec accumulator using V_WMMA_F32_16X16X4_F32.
// Wave owns 16 output rows R..R+15. Lane = (h = lane>>4, lrow = lane&15).
// Lane streams W row (R+lrow), columns k+4h..k+4h+3 per 8-wide K chunk
// (b128 load). The vector x is replicated across all 16 B columns, so every
// column of the 16x16 D tile holds the same dot product for its row.
// K-chunk-to-WMMA mapping is a permutation applied identically to A and B,
// which is legal for a dot product.
// ---------------------------------------------------------------------------
__device__ __forceinline__ v8f wmma_dot16(const float* __restrict__ Wrow,
                                          const float* __restrict__ x,
                                          int K, int h, v8f c) {
  const float* wp = Wrow + 4 * h;
  const float* xp = x + 4 * h;
#pragma unroll 4
  for (int k = 0; k < K; k += 8) {
    v4f wv = *(const v4f*)(wp + k);
    v4f xv = *(const v4f*)(xp + k);
    v2f a0 = {wv.x, wv.y}; v2f b0 = {xv.x, xv.y};
    v2f a1 = {wv.z, wv.w}; v2f b1 = {xv.z, xv.w};
    c = __builtin_amdgcn_wmma_f32_16x16x4_f32(false, a0, false, b0, (short)0, c, false, false);
    c = __builtin_amdgcn_wmma_f32_16x16x4_f32(false, a1, false, b1, (short)0, c, false, false);
  }
  return c;
}

// D-tile extraction: VGPR v holds M=v (lanes 0-15) and M=v+8 (lanes 16-31);
// all N columns identical -> lanes with lrow==0 in each half own 8 rows each.

// ---- Kernel 0: x = relu(emb[idx]) -----------------------------------------
__global__ void k_embed(const int* __restrict__ idx, const float* __restrict__ emb,
                        float* __restrict__ ws) {
  int i = blockIdx.x * blockDim.x + threadIdx.x;
  if (i < E_SIZE) {
    float v = emb[(long)idx[0] * E_SIZE + i];
    ws[WS_X + i] = v > 0.0f ? v : 0.0f;
  }
}

// ---- Kernel 1: gates = W_ih@x + W_hh@h0 + b_ih + b_hh  (4096 rows) --------
__global__ void k_gates(const float* __restrict__ W_ih, const float* __restrict__ W_hh,
                        const float* __restrict__ b_ih, const float* __restrict__ b_hh,
                        const float* __restrict__ h0, float* __restrict__ ws) {
  int tid = threadIdx.x;
  int lane = tid & 31, wave = tid >> 5;
  int h = lane >> 4, lrow = lane & 15;
  int R = (blockIdx.x * 8 + wave) * 16;     // 32 blocks * 8 waves = 256 tiles
  v8f c = {0, 0, 0, 0, 0, 0, 0, 0};
  c = wmma_dot16(W_ih + (long)(R + lrow) * E_SIZE, ws + WS_X, E_SIZE, h, c);
  c = wmma_dot16(W_hh + (long)(R + lrow) * H_SIZE, h0, H_SIZE, h, c);
  if (lrow == 0) {
    int base = R + 8 * h;
#pragma unroll
    for (int v = 0; v < 8; ++v) {
      int r = base + v;
      ws[WS_GATES + r] = c[v] + b_ih[r] + b_hh[r];
    }
  }
}

// ---- Kernel 2: LSTM cell elementwise --------------------------------------
__global__ void k_cell(const float* __restrict__ c0, float* __restrict__ ws,
                       float* __restrict__ out) {
  int e = blockIdx.x * blockDim.x + threadIdx.x;
  if (e >= H_SIZE) return;
  float gi = ws[WS_GATES + e];
  float gf = ws[WS_GATES + H_SIZE + e];
  float gg = ws[WS_GATES + 2 * H_SIZE + e];
  float go = ws[WS_GATES + 3 * H_SIZE + e];
  float cn = sigmoidf_(gf) * c0[e] + sigmoidf_(gi) * tanhf(gg);
  float hn = sigmoidf_(go) * tanhf(cn);
  ws[WS_CNEW + e] = cn;
  ws[WS_HNEW + e] = hn;
  out[V_SIZE + e] = hn;                // h output
  out[V_SIZE + H_SIZE + e] = cn;       // c output
}

// ---- Kernel 3: a[s] = dot(hs[s], h_new)  (4096 rows) ----------------------
__global__ void k_scores(const float* __restrict__ hs, float* __restrict__ ws) {
  int tid = threadIdx.x;
  int lane = tid & 31, wave = tid >> 5;
  int h = lane >> 4, lrow = lane & 15;
  int R = (blockIdx.x * 8 + wave) * 16;
  v8f c = {0, 0, 0, 0, 0, 0, 0, 0};
  c = wmma_dot16(hs + (long)(R + lrow) * H_SIZE, ws + WS_HNEW, H_SIZE, h, c);
  if (lrow == 0) {
    int base = R + 8 * h;
#pragma unroll
    for (int v = 0; v < 8; ++v) ws[WS_ASCORE + base + v] = c[v];
  }
}

// ---- Kernel 4: partial ctx: block b sums 64 rows of a[s]*hs[s] ------------
__global__ void k_ctx_partial(const float* __restrict__ hs, float* __restrict__ ws) {
  int b = blockIdx.x, tid = threadIdx.x;
  float acc[4] = {0.f, 0.f, 0.f, 0.f};
  int s0 = b * 64;
  for (int s = s0; s < s0 + 64; ++s) {
    float av = ws[WS_ASCORE + s];
    const float* row = hs + (long)s * H_SIZE;
#pragma unroll
    for (int j = 0; j < 4; ++j) acc[j] += av * row[tid + 256 * j];  // coalesced
  }
#pragma unroll
  for (int j = 0; j < 4; ++j) ws[WS_CTXPART + b * H_SIZE + tid + 256 * j] = acc[j];
}

// ---- Kernel 5: reduce partials -> ctx, build cat = [ctx ; h_new] ----------
__global__ void k_ctx_reduce(float* __restrict__ ws) {
  int e = blockIdx.x * 256 + threadIdx.x;   // 4 blocks * 256 = 1024
  float s = 0.f;
  for (int b = 0; b < 64; ++b) s += ws[WS_CTXPART + b * H_SIZE + e];
  float ctx = s * (1.0f / (float)S_SIZE);
  ws[WS_CAT + e] = ctx;
  ws[WS_CAT + H_SIZE + e] = ws[WS_HNEW + e];
}

// ---- Kernel 6: final_hidden = tanh(W_final@cat + b_final) (1024 rows, K=2048)
__global__ void k_final(const float* __restrict__ W_final, const float* __restrict__ b_final,
                        float* __restrict__ ws, float* __restrict__ out) {
  int tid = threadIdx.x;
  int lane = tid & 31, wave = tid >> 5;
  int h = lane >> 4, lrow = lane & 15;
  int R = (blockIdx.x * 8 + wave) * 16;     // 8 blocks -> 64 tiles -> 1024 rows
  v8f c = {0, 0, 0, 0, 0, 0, 0, 0};
  c = wmma_dot16(W_final + (long)(R + lrow) * (2 * H_SIZE), ws + WS_CAT, 2 * H_SIZE, h, c);
  if (lrow == 0) {
    int base = R + 8 * h;
#pragma unroll
    for (int v = 0; v < 8; ++v) {
      int r = base + v;
      float fh = tanhf(c[v] + b_final[r]);
      ws[WS_FINAL + r] = fh;
      out[V_SIZE + 2 * H_SIZE + r] = fh;     // final_hidden output
    }
  }
}

// ---- Kernel 7: mid = W_out@final + b_out  (50257 rows, dominant 205 MB) ---
__global__ void k_out(const float* __restrict__ W_out, const float* __restrict__ b_out,
                      float* __restrict__ ws) {
  int tid = threadIdx.x;
  int lane = tid & 31, wave = tid >> 5;
  int h = lane >> 4, lrow = lane & 15;
  int tile = blockIdx.x * 8 + wave;          // 3142 tiles; wave-uniform guard
  if (tile >= (V_SIZE + 15) / 16) return;
  int R = tile * 16;
  int row = R + lrow;
  if (row > V_SIZE - 1) row = V_SIZE - 1;    // clamp loads; EXEC stays all-1s
  v8f c = {0, 0, 0, 0, 0, 0, 0, 0};
  c = wmma_dot16(W_out + (long)row * H_SIZE, ws + WS_FINAL, H_SIZE, h, c);
  if (lrow == 0) {
    int base = R + 8 * h;
#pragma unroll
    for (int v = 0; v < 8; ++v) {
      int r = base + v;
      if (r < V_SIZE) ws[WS_MID + r] = c[v] + b_out[r];
    }
  }
}

// ---- Kernel 8: per-block online (max, sumexp) over mid --------------------
__global__ void k_stats(float* __restrict__ ws) {
  __shared__ float sm[256], ss[256];
  int tid = threadIdx.x;
  float m = -3.402823466e38f, s = 0.f;
  for (int i = blockIdx.x * 256 + tid; i < V_SIZE; i += 64 * 256) {
    float v = ws[WS_MID + i];
    if (v > m) { s = s * expf(m - v) + 1.f; m = v; }
    else       { s += expf(v - m); }
  }
  sm[tid] = m; ss[tid] = s;
  __syncthreads();
  for (int off = 128; off > 0; off >>= 1) {
    if (tid < off) {
      float m2 = sm[tid + off], s2 = ss[tid + off];
      float M = fmaxf(sm[tid], m2);
      ss[tid] = ss[tid] * expf(sm[tid] - M) + s2 * expf(m2 - M);
      sm[tid] = M;
    }
    __syncthreads();
  }
  if (tid == 0) {
    ws[WS_STATS + blockIdx.x * 2]     = sm[0];
    ws[WS_STATS + blockIdx.x * 2 + 1] = ss[0];
  }
}

// ---- Kernel 9: combine 64 partials -> logZ --------------------------------
__global__ void k_combine(float* __restrict__ ws) {
  if (threadIdx.x == 0) {
    float M = -3.402823466e38f, S = 0.f;
    for (int b = 0; b < 64; ++b) {
      float m2 = ws[WS_STATS + 2 * b], s2 = ws[WS_STATS + 2 * b + 1];
      float Mn = fmaxf(M, m2);
      S = S * expf(M - Mn) + s2 * expf(m2 - Mn);
      M = Mn;
    }
    ws[WS_LOGZ] = M + logf(S);
  }
}

// ---- Kernel 10: output = mid - logZ ---------------------------------------
__global__ void k_write(float* __restrict__ out, const float* __restrict__ ws) {
  float lz = ws[WS_LOGZ];
  for (int i = blockIdx.x * 256 + threadIdx.x; i < V_SIZE; i += 64 * 256)
    out[i] = ws[WS_MID + i] - lz;
}

extern "C" void kernel_launch(void* const* d_in, const int* in_sizes, int n_in,
                              void* d_out, int out_size, void* d_ws, size_t ws_size,
                              hipStream_t stream) {
  const int*   idx   = (const int*)  d_in[0];
  const float* h0    = (const float*)d_in[1];
  const float* c0    = (const float*)d_in[2];
  const float* hs    = (const float*)d_in[3];   // input_hidden (4096,1,1024)
  const float* emb   = (const float*)d_in[4];
  const float* W_ih  = (const float*)d_in[5];
  const float* W_hh  = (const float*)d_in[6];
  const float* b_ih  = (const float*)d_in[7];
  const float* b_hh  = (const float*)d_in[8];
  const float* W_fin = (const float*)d_in[9];
  const float* b_fin = (const float*)d_in[10];
  const float* W_out = (const float*)d_in[11];
  const float* b_out = (const float*)d_in[12];
  float* out = (float*)d_out;
  float* ws  = (float*)d_ws;   // requires ~522 KB

  k_embed      <<<4,   256, 0, stream>>>(idx, emb, ws);
  k_gates      <<<32,  256, 0, stream>>>(W_ih, W_hh, b_ih, b_hh, h0, ws);
  k_cell       <<<4,   256, 0, stream>>>(c0, ws, out);
  k_scores     <<<32,  256, 0, stream>>>(hs, ws);
  k_ctx_partial<<<64,  256, 0, stream>>>(hs, ws);
  k_ctx_reduce <<<4,   256, 0, stream>>>(ws);
  k_final      <<<8,   256, 0, stream>>>(W_fin, b_fin, ws, out);
  k_out        <<<393, 256, 0, stream>>>(W_out, b_out, ws);
  k_stats      <<<64,  256, 0, stream>>>(ws);
  k_combine    <<<1,   64,  0, stream>>>(ws);
  k_write      <<<64,  256, 0, stream>>>(out, ws);
}